// IDHT_45002667327994
// MI455X (gfx1250) — compile-verified
//
#include <hip/hip_runtime.h>
#include <hip/hip_bf16.h>

// IDHT (inverse deep Hough transform) via one-hot WMMA on gfx1250.
//
// out[nc, h, w] = sum_a hough[nc, a, idx(a,h,w)]
// idx(a,h,w) = rint(cos(th_a)*(w-128) + sin(th_a)*(h-128)) + 361   (never clips:
// |r| <= 128*sqrt(2) ~ 181 < 361, so idx in [179, 543] always in-range).
//
// For fixed (a, h) and 16 consecutive w, idx is monotone with span <= 15,
// so the gather is a dense 16-wide window [r0, r0+15] + a one-hot 16x16
// selection matrix. Per angle:
//   D(16nc x 16pix) += sum_{j=0..3} A_j(16nc x 4) * B_j(4 x 16)
// using V_WMMA_F32_16X16X4_F32 (exact fp32 — one-hot B is exact).
// Two independent accumulator chains (even/odd j) keep 2 WMMAs in flight
// per wave instead of one serial D->C chain of 720.

typedef __attribute__((ext_vector_type(2))) float v2f;
typedef __attribute__((ext_vector_type(8))) float v8f;

#define A_ANG 180
#define NRHO  723
#define H_OUT 256
#define W_OUT 256
#define HW    (H_OUT * W_OUT)

__global__ __launch_bounds__(256) void idht_wmma_kernel(
    const float* __restrict__ hough,   // [512, 180, 723]  (N*C flattened)
    float* __restrict__ out)           // [512, 256, 256]
{
    // f64 trig table (matches numpy's float64 cos/sin + rint convention)
    __shared__ double cs[2 * A_ANG];
    const int tid = threadIdx.x;
    if (tid < A_ANG) {
        double th = (double)tid * (3.14159265358979323846 / (double)A_ANG);
        cs[2 * tid]     = cos(th);
        cs[2 * tid + 1] = sin(th);
    }
    __syncthreads();

    const int lane = tid & 31;
    const int wave = tid >> 5;
    const int m    = lane & 15;   // A-operand: nc row in tile; B/D: pixel column
    const int hi   = lane >> 4;   // lane half -> K rows {0,1} vs {2,3}, D rows v vs v+8

    // block -> (pixel-group, nc-quad); wave -> nc tile of 16
    const int pg     = blockIdx.x >> 2;          // 0..4095 : h*16 + w-group
    const int q      = blockIdx.x & 3;           // 0..3
    const int ncTile = q * 8 + wave;             // 0..31
    const int ncBase = ncTile << 4;              // 0..496

    const int h  = pg >> 4;
    const int w0 = (pg & 15) << 4;

    const double xd = (double)(w0 + m - (W_OUT / 2));  // this lane's pixel x
    const double yd = (double)(h - (H_OUT / 2));

    // per-lane base row: hough[ncBase+m, :, :]
    const float* rowA = hough + (size_t)(ncBase + m) * (A_ANG * NRHO);

    v8f acc0 = {};
    v8f acc1 = {};

    for (int a = 0; a < A_ANG; ++a) {
        const double c = cs[2 * a];
        const double s = cs[2 * a + 1];

        // exact-as-numpy rho bin for this lane's pixel (f64 fma + rndne)
        const int idx = (int)rint(c * xd + s * yd) + (NRHO / 2);

        // idx is monotone over the 16 pixels -> window base = min of endpoints.
        // readlane makes r0 an SGPR (scalar window base for the A loads).
        const int i0  = __builtin_amdgcn_readlane(idx, 0);
        const int i15 = __builtin_amdgcn_readlane(idx, 15);
        const int r0  = (i0 < i15) ? i0 : i15;
        const int d   = idx - r0;   // 0..15 : one-hot position of this pixel

        // dense window loads: lane covers columns (4j + 2*hi, 4j + 2*hi + 1)
        const float* rw = rowA + a * NRHO + r0 + 2 * hi;

#pragma unroll
        for (int j = 0; j < 4; ++j) {
            v2f Aj;
            Aj.x = rw[4 * j];
            Aj.y = rw[4 * j + 1];

            const int kbase = 4 * j + 2 * hi;
            v2f Bj;
            Bj.x = (d == kbase)     ? 1.0f : 0.0f;
            Bj.y = (d == kbase + 1) ? 1.0f : 0.0f;

            // D = A x B + C, fp32 16x16x4; alternate accumulator chains
            if (j & 1) {
                acc1 = __builtin_amdgcn_wmma_f32_16x16x4_f32(
                    false, Aj, false, Bj, (short)0, acc1, false, false);
            } else {
                acc0 = __builtin_amdgcn_wmma_f32_16x16x4_f32(
                    false, Aj, false, Bj, (short)0, acc0, false, false);
            }
        }
    }

    // D layout: VGPR v -> nc row v (lanes 0-15) / v+8 (lanes 16-31), N = lane&15
    const size_t pixoff = (size_t)h * W_OUT + (size_t)(w0 + m);
#pragma unroll
    for (int v = 0; v < 8; ++v) {
        out[(size_t)(ncBase + v + 8 * hi) * HW + pixoff] = acc0[v] + acc1[v];
    }
}

extern "C" void kernel_launch(void* const* d_in, const int* in_sizes, int n_in,
                              void* d_out, int out_size, void* d_ws, size_t ws_size,
                              hipStream_t stream) {
    const float* hough = (const float*)d_in[0];
    float* out = (float*)d_out;
    (void)in_sizes; (void)n_in; (void)out_size; (void)d_ws; (void)ws_size;

    // 4096 pixel groups (256 rows x 16 groups of 16 w) x 4 nc-quads
    dim3 grid(16384);
    dim3 block(256);  // 8 waves: each wave = 16 nc x 16 pixels
    hipLaunchKernelGGL(idht_wmma_kernel, grid, block, 0, stream, hough, out);
}